// RPNSingleModule_43662637532142
// MI455X (gfx1250) — compile-verified
//
#include <hip/hip_runtime.h>
#include <hip/hip_bf16.h>
#include <math.h>

typedef float v2f __attribute__((ext_vector_type(2)));
typedef float v8f __attribute__((ext_vector_type(8)));

#define BN   4
#define INC_ 256
#define OUTC_ 256
#define HH   64
#define WW   96
#define HWPX (HH*WW)          // 6144
#define AA   9
#define NANCH (HWPX*AA)       // 55296
#define NPAD  65536
#define PRE_NMS_ 6000
#define POST_NMS_ 1000
#define NMS_TH_ 0.7f
#define KTOT (INC_*9)         // 2304

// ---------------- device helpers ----------------

__device__ __forceinline__ void anchor_at(int i, float& ax1, float& ay1,
                                          float& ax2, float& ay2) {
  int a = i % AA;
  int pos = i / AA;
  int gx = pos % WW;
  int gy = pos / WW;
  int ri = a / 3, si = a % 3;
  const float ratios[3] = {0.5f, 1.0f, 2.0f};
  const float scales[3] = {8.0f, 16.0f, 32.0f};
  float r = ratios[ri], s = scales[si];
  float wsr = rintf(sqrtf(256.0f / r));   // np.round == round-half-even == rintf
  float hsr = rintf(wsr * r);
  float w2 = wsr * s, h2 = hsr * s;
  float cx = 7.5f + gx * 16.0f, cy = 7.5f + gy * 16.0f;
  ax1 = cx - 0.5f * (w2 - 1.0f);
  ay1 = cy - 0.5f * (h2 - 1.0f);
  ax2 = cx + 0.5f * (w2 - 1.0f);
  ay2 = cy + 0.5f * (h2 - 1.0f);
}

__device__ __forceinline__ float iou1(float ax1, float ay1, float ax2, float ay2,
                                      float bx1, float by1, float bx2, float by2) {
  float aa = (ax2 - ax1 + 1.0f) * (ay2 - ay1 + 1.0f);
  float ab = (bx2 - bx1 + 1.0f) * (by2 - by1 + 1.0f);
  float x1 = fmaxf(ax1, bx1), y1 = fmaxf(ay1, by1);
  float x2 = fminf(ax2, bx2), y2 = fminf(ay2, by2);
  float w = fmaxf(x2 - x1 + 1.0f, 0.0f), h = fmaxf(y2 - y1 + 1.0f, 0.0f);
  float inter = w * h;
  return inter / (aa + ab - inter);
}

// ---------------- kernel 1: 3x3 conv + bias + relu via f32 WMMA ----------------
// K order permuted to tap-major (k = tap*256 + ic) so per-step address math is
// a pure linear walk; boundary handling is a loop-invariant mask per tap
// (clamped addresses, no EXEC divergence). Each wave owns a 16outC x 32px tile
// (two accumulators sharing one A operand). 8 waves/block share (b, mtile) and
// a 16KB LDS weight slab per tap.

__global__ void conv3x3_wmma(const float* __restrict__ x,
                             const float* __restrict__ w_pre,
                             const float* __restrict__ b_pre,
                             float* __restrict__ feat) {
  __shared__ float sw[16 * 256];   // 16 KB: [ocl][ic] for current tap
  const int tile = blockIdx.x * 8 + (threadIdx.x >> 5);
  const int lane = threadIdx.x & 31;
  const int b  = tile / (16 * 192);
  const int mt = (tile / 192) % 16;
  const int nt = tile % 192;
  const int y  = nt / 3;
  const int x0 = (nt % 3) * 32;
  const int half = lane >> 4;
  const int mrow = lane & 15;      // M for A reads
  const int ncol = lane & 15;      // N for B reads / C stores
  const int ko   = half * 2;       // K sub-offset for this lane half

  v8f c0 = {};
  v8f c1 = {};
  const float* xb = x + (size_t)b * INC_ * HWPX;

  for (int r = 0; r < 9; ++r) {
    // stage permuted weights for tap r: sw[ocl*256+ic] = w[oc][ic][r]
    for (int e = threadIdx.x; e < 16 * 256; e += 256) {
      int ocl = e >> 8;
      int ic  = e & 255;
      sw[e] = w_pre[(size_t)(mt * 16 + ocl) * KTOT + ic * 9 + r];
    }
    __syncthreads();

    const int dy = r / 3 - 1, dx = r % 3 - 1;
    const int yy = y + dy;
    const bool vy = (yy >= 0) && (yy < HH);
    const int yyc = yy < 0 ? 0 : (yy > HH - 1 ? HH - 1 : yy);
    const int xa = x0 + ncol + dx;
    const int xb2 = xa + 16;
    const float ma = (vy && xa >= 0 && xa < WW) ? 1.0f : 0.0f;
    const float mb = (vy && xb2 >= 0 && xb2 < WW) ? 1.0f : 0.0f;
    const int xac = xa < 0 ? 0 : (xa > WW - 1 ? WW - 1 : xa);
    const int xbc = xb2 < 0 ? 0 : (xb2 > WW - 1 ? WW - 1 : xb2);
    const float* pa = xb + yyc * WW + xac;
    const float* pb = xb + yyc * WW + xbc;

#pragma unroll 4
    for (int k = 0; k < 256; k += 4) {
      v2f a, b0, b1;
      a.x = sw[mrow * 256 + k + ko + 0];
      a.y = sw[mrow * 256 + k + ko + 1];
      b0.x = ma * pa[(size_t)(k + ko + 0) * HWPX];
      b0.y = ma * pa[(size_t)(k + ko + 1) * HWPX];
      b1.x = mb * pb[(size_t)(k + ko + 0) * HWPX];
      b1.y = mb * pb[(size_t)(k + ko + 1) * HWPX];
      c0 = __builtin_amdgcn_wmma_f32_16x16x4_f32(false, a, false, b0,
                                                 (short)0, c0, false, false);
      c1 = __builtin_amdgcn_wmma_f32_16x16x4_f32(false, a, false, b1,
                                                 (short)0, c1, false, false);
    }
    __syncthreads();
  }

#pragma unroll
  for (int rr = 0; rr < 8; ++rr) {
    int M = rr + half * 8;
    int oc = mt * 16 + M;
    float bia = b_pre[oc];
    float v0 = c0[rr] + bia;
    float v1 = c1[rr] + bia;
    v0 = v0 > 0.0f ? v0 : 0.0f;
    v1 = v1 > 0.0f ? v1 : 0.0f;
    size_t rowbase = ((size_t)(b * OUTC_ + oc) * HH + y) * WW;
    feat[rowbase + x0 + ncol] = v0;
    feat[rowbase + x0 + 16 + ncol] = v1;
  }
}

// ---------------- kernel 2: 1x1 head convs (cls 18ch, box 36ch) ----------------

__global__ void conv1x1_heads(const float* __restrict__ feat,
                              const float* __restrict__ w_cls,
                              const float* __restrict__ b_cls,
                              const float* __restrict__ w_box,
                              const float* __restrict__ b_box,
                              float* __restrict__ scoresb,
                              float* __restrict__ deltasb) {
  int t = blockIdx.x * blockDim.x + threadIdx.x;
  if (t >= BN * 54 * HWPX) return;
  int pos = t % HWPX;
  int rest = t / HWPX;
  int c = rest % 54;
  int b = rest / 54;
  const float* fb = feat + (size_t)b * OUTC_ * HWPX + pos;
  float acc;
  const float* wv;
  if (c < 18) { wv = w_cls + c * OUTC_; acc = b_cls[c]; }
  else        { wv = w_box + (c - 18) * OUTC_; acc = b_box[c - 18]; }
  for (int m = 0; m < OUTC_; ++m)
    acc += wv[m] * fb[(size_t)m * HWPX];
  if (c < 18) scoresb[((size_t)b * 18 + c) * HWPX + pos] = acc;
  else        deltasb[((size_t)b * 36 + (c - 18)) * HWPX + pos] = acc;
}

// ---------------- kernel 3: proposal scores/boxes + sort keys ----------------

__global__ void proposals(const float* __restrict__ scoresb,
                          const float* __restrict__ deltasb,
                          const float* __restrict__ im_info,
                          float* __restrict__ pbox,
                          float* __restrict__ pscore,
                          unsigned long long* __restrict__ keys) {
  int t = blockIdx.x * blockDim.x + threadIdx.x;
  if (t >= BN * NPAD) return;
  int b = t / NPAD;
  int i = t % NPAD;
  unsigned long long* kb = keys + (size_t)b * NPAD;
  if (i >= NANCH) { kb[i] = 0ull; return; }
  int a = i % AA;
  int pos = i / AA;
  float s0 = scoresb[((size_t)b * 18 + 2 * a + 0) * HWPX + pos];
  float s1 = scoresb[((size_t)b * 18 + 2 * a + 1) * HWPX + pos];
  float fg = 1.0f / (1.0f + expf(s0 - s1));   // softmax fg prob
  float d0 = deltasb[((size_t)b * 36 + 4 * a + 0) * HWPX + pos];
  float d1 = deltasb[((size_t)b * 36 + 4 * a + 1) * HWPX + pos];
  float d2 = deltasb[((size_t)b * 36 + 4 * a + 2) * HWPX + pos];
  float d3 = deltasb[((size_t)b * 36 + 4 * a + 3) * HWPX + pos];
  float ax1, ay1, ax2, ay2;
  anchor_at(i, ax1, ay1, ax2, ay2);
  float wa = ax2 - ax1 + 1.0f, ha = ay2 - ay1 + 1.0f;
  float cxa = ax1 + 0.5f * wa, cya = ay1 + 0.5f * ha;
  float cx = d0 * wa + cxa, cy = d1 * ha + cya;
  float pw = expf(d2) * wa, ph = expf(d3) * ha;
  float ih = im_info[b * 3 + 0], iw = im_info[b * 3 + 1], sc = im_info[b * 3 + 2];
  float x1 = fminf(fmaxf(cx - 0.5f * pw, 0.0f), iw - 1.0f);
  float y1 = fminf(fmaxf(cy - 0.5f * ph, 0.0f), ih - 1.0f);
  float x2 = fminf(fmaxf(cx + 0.5f * pw, 0.0f), iw - 1.0f);
  float y2 = fminf(fmaxf(cy + 0.5f * ph, 0.0f), ih - 1.0f);
  bool ok = ((x2 - x1 + 1.0f) >= 16.0f * sc) && ((y2 - y1 + 1.0f) >= 16.0f * sc);
  float scv = ok ? fg : -INFINITY;
  float* pb = pbox + ((size_t)b * NANCH + i) * 4;
  pb[0] = x1; pb[1] = y1; pb[2] = x2; pb[3] = y2;
  pscore[(size_t)b * NANCH + i] = scv;
  unsigned u = __float_as_uint(scv);
  unsigned mono = (u & 0x80000000u) ? ~u : (u | 0x80000000u);
  kb[i] = ((unsigned long long)mono << 32) | (unsigned)(~(unsigned)i);
}

// ---------------- kernel 4: per-image bitonic sort (descending) ----------------

__global__ void bitonic_sort(unsigned long long* __restrict__ keys) {
  unsigned long long* k = keys + (size_t)blockIdx.x * NPAD;
  for (unsigned int len = 2; len <= NPAD; len <<= 1) {
    for (unsigned int j = len >> 1; j > 0; j >>= 1) {
      for (unsigned int idx = threadIdx.x; idx < NPAD; idx += blockDim.x) {
        unsigned int p = idx ^ j;
        if (p > idx) {
          bool desc = ((idx & len) == 0);
          unsigned long long va = k[idx], vb = k[p];
          bool sw = desc ? (va < vb) : (va > vb);
          if (sw) { k[idx] = vb; k[p] = va; }
        }
      }
      __syncthreads();
    }
  }
}

// ---------------- kernel 5: greedy NMS + roi output ----------------

__global__ void nms_rois(const unsigned long long* __restrict__ keys,
                         const float* __restrict__ pbox,
                         const float* __restrict__ pscore,
                         float* __restrict__ tbox,
                         float* __restrict__ tsc,
                         float* __restrict__ rois) {
  const int b = blockIdx.x;
  const int tid = threadIdx.x;
  __shared__ unsigned char sup[PRE_NMS_];
  __shared__ int keepl[POST_NMS_];
  __shared__ int s_kept;
  __shared__ int s_stop;
  float* tb = tbox + (size_t)b * PRE_NMS_ * 4;
  float* ts = tsc + (size_t)b * PRE_NMS_;
  for (int j = tid; j < PRE_NMS_; j += blockDim.x) {
    unsigned long long kk = keys[(size_t)b * NPAD + j];
    unsigned idx = ~(unsigned)(kk & 0xFFFFFFFFull);
    const float* pb = pbox + ((size_t)b * NANCH + idx) * 4;
    tb[j * 4 + 0] = pb[0]; tb[j * 4 + 1] = pb[1];
    tb[j * 4 + 2] = pb[2]; tb[j * 4 + 3] = pb[3];
    ts[j] = pscore[(size_t)b * NANCH + idx];
    sup[j] = 0;
  }
  if (tid == 0) { s_kept = 0; s_stop = 0; }
  __syncthreads();
  for (int cur = 0; cur < PRE_NMS_; ++cur) {
    __syncthreads();
    if (s_stop || s_kept >= POST_NMS_) break;
    if (sup[cur]) continue;
    float scc = ts[cur];
    if (scc <= -1e30f) { if (tid == 0) s_stop = 1; continue; }
    if (tid == 0) keepl[s_kept] = cur;
    float bx1 = tb[cur * 4 + 0], by1 = tb[cur * 4 + 1];
    float bx2 = tb[cur * 4 + 2], by2 = tb[cur * 4 + 3];
    for (int j = cur + 1 + tid; j < PRE_NMS_; j += blockDim.x) {
      if (!sup[j]) {
        float ov = iou1(bx1, by1, bx2, by2,
                        tb[j * 4 + 0], tb[j * 4 + 1], tb[j * 4 + 2], tb[j * 4 + 3]);
        if (ov > NMS_TH_) sup[j] = 1;
      }
    }
    __syncthreads();
    if (tid == 0) s_kept++;
  }
  __syncthreads();
  for (int s = tid; s < POST_NMS_; s += blockDim.x) {
    float* rr = rois + ((size_t)b * POST_NMS_ + s) * 5;
    rr[0] = (float)b;
    if (s < s_kept) {
      int j = keepl[s];
      rr[1] = tb[j * 4 + 0]; rr[2] = tb[j * 4 + 1];
      rr[3] = tb[j * 4 + 2]; rr[4] = tb[j * 4 + 3];
    } else {
      rr[1] = 0.0f; rr[2] = 0.0f; rr[3] = 0.0f; rr[4] = 0.0f;
    }
  }
}

// ---------------- kernel 6: accumulator init ----------------

__global__ void init_acc(unsigned* gtmax, float* acc, int* cnt) {
  int t = threadIdx.x;
  if (t < BN * 20) gtmax[t] = 0u;       // bits of 0.0f
  if (t < 1 + BN) acc[t] = 0.0f;        // acc[0]=ce_sum, acc[1..4]=sm_sum[b]
  if (t < BN) cnt[t] = 0;
}

// ---------------- kernel 7: per-anchor IoU stats ----------------

__global__ void iou_stats(const float* __restrict__ gt_boxes,
                          const float* __restrict__ im_info,
                          float* __restrict__ maxov,
                          int* __restrict__ argmx,
                          unsigned* __restrict__ gtmax) {
  int t = blockIdx.x * blockDim.x + threadIdx.x;
  if (t >= BN * NANCH) return;
  int b = t / NANCH;
  int i = t % NANCH;
  float ax1, ay1, ax2, ay2;
  anchor_at(i, ax1, ay1, ax2, ay2);
  float ih = im_info[b * 3 + 0], iw = im_info[b * 3 + 1];
  bool inside = (ax1 >= 0.0f) && (ay1 >= 0.0f) && (ax2 < iw) && (ay2 < ih);
  float mx = 0.0f;
  int am = 0;
  for (int g = 0; g < 20; ++g) {
    const float* gb = gt_boxes + ((size_t)b * 20 + g) * 5;
    float v = inside ? iou1(ax1, ay1, ax2, ay2, gb[0], gb[1], gb[2], gb[3]) : 0.0f;
    if (v > mx) { mx = v; am = g; }
    atomicMax(&gtmax[b * 20 + g], __float_as_uint(v));
  }
  maxov[t] = mx;
  argmx[t] = am;
}

// ---------------- kernel 8: label assignment ----------------

__global__ void assign_labels(const float* __restrict__ gt_boxes,
                              const float* __restrict__ im_info,
                              const float* __restrict__ maxov,
                              const unsigned* __restrict__ gtmax,
                              int* __restrict__ labels) {
  int t = blockIdx.x * blockDim.x + threadIdx.x;
  if (t >= BN * NANCH) return;
  int b = t / NANCH;
  int i = t % NANCH;
  float ax1, ay1, ax2, ay2;
  anchor_at(i, ax1, ay1, ax2, ay2);
  float ih = im_info[b * 3 + 0], iw = im_info[b * 3 + 1];
  bool inside = (ax1 >= 0.0f) && (ay1 >= 0.0f) && (ax2 < iw) && (ay2 < ih);
  int lab = -1;
  if (inside) {
    float mx = maxov[t];
    if (mx < 0.3f) lab = 0;
    bool best = false;
    for (int g = 0; g < 20; ++g) {
      const float* gb = gt_boxes + ((size_t)b * 20 + g) * 5;
      float v = iou1(ax1, ay1, ax2, ay2, gb[0], gb[1], gb[2], gb[3]);
      float gm = __uint_as_float(gtmax[b * 20 + g]);
      if (v == gm && gm > 0.0f) best = true;
    }
    if (best) lab = 1;
    if (mx >= 0.7f) lab = 1;
  }
  labels[t] = lab;
}

// ---------------- kernel 9: deterministic subsampling (cumsum ranks) ----------

__global__ void subsample(int* __restrict__ labels) {
  const int b = blockIdx.x;
  int* L = labels + (size_t)b * NANCH;
  __shared__ int part[1024];
  __shared__ int s_total;
  const int tid = threadIdx.x;
  const int seg = NANCH / 1024;   // 54
  const int st = tid * seg;
  // FG pass
  int loc = 0;
  for (int e = 0; e < seg; ++e) loc += (L[st + e] == 1) ? 1 : 0;
  part[tid] = loc;
  __syncthreads();
  if (tid == 0) {
    int run = 0;
    for (int q = 0; q < 1024; ++q) { int v = part[q]; part[q] = run; run += v; }
    s_total = run;
  }
  __syncthreads();
  int rank = part[tid];
  for (int e = 0; e < seg; ++e) {
    int idx = st + e;
    if (L[idx] == 1) { if (rank >= 128) L[idx] = -1; rank++; }
  }
  int n_fg = s_total < 128 ? s_total : 128;
  __syncthreads();
  // BG pass
  loc = 0;
  for (int e = 0; e < seg; ++e) loc += (L[st + e] == 0) ? 1 : 0;
  part[tid] = loc;
  __syncthreads();
  if (tid == 0) {
    int run = 0;
    for (int q = 0; q < 1024; ++q) { int v = part[q]; part[q] = run; run += v; }
  }
  __syncthreads();
  rank = part[tid];
  int cap = 256 - n_fg;
  for (int e = 0; e < seg; ++e) {
    int idx = st + e;
    if (L[idx] == 0) { if (rank >= cap) L[idx] = -1; rank++; }
  }
}

// ---------------- kernel 10: loss accumulation ----------------

__global__ void loss_accum(const float* __restrict__ scoresb,
                           const float* __restrict__ deltasb,
                           const float* __restrict__ gt_boxes,
                           const int* __restrict__ labels,
                           const int* __restrict__ argmx,
                           float* __restrict__ acc,
                           int* __restrict__ cnt) {
  int t = blockIdx.x * blockDim.x + threadIdx.x;
  if (t >= BN * NANCH) return;
  int b = t / NANCH;
  int i = t % NANCH;
  int l = labels[t];
  if (l < 0) return;
  int a = i % AA;
  int pos = i / AA;
  float s0 = scoresb[((size_t)b * 18 + 2 * a + 0) * HWPX + pos];
  float s1 = scoresb[((size_t)b * 18 + 2 * a + 1) * HWPX + pos];
  float m = fmaxf(s0, s1);
  float lse = m + logf(expf(s0 - m) + expf(s1 - m));
  float sl = (l == 1) ? s1 : s0;
  atomicAdd(&acc[0], lse - sl);
  atomicAdd(&cnt[b], 1);
  if (l == 1) {
    float ax1, ay1, ax2, ay2;
    anchor_at(i, ax1, ay1, ax2, ay2);
    float wa = ax2 - ax1 + 1.0f, ha = ay2 - ay1 + 1.0f;
    float cxa = ax1 + 0.5f * wa, cya = ay1 + 0.5f * ha;
    const float* gb = gt_boxes + ((size_t)b * 20 + argmx[t]) * 5;
    float wg = gb[2] - gb[0] + 1.0f, hg = gb[3] - gb[1] + 1.0f;
    float cxg = gb[0] + 0.5f * wg, cyg = gb[1] + 0.5f * hg;
    float tg[4];
    tg[0] = (cxg - cxa) / wa;
    tg[1] = (cyg - cya) / ha;
    tg[2] = logf(wg / wa);
    tg[3] = logf(hg / ha);
    float sm = 0.0f;
    for (int d = 0; d < 4; ++d) {
      float pred = deltasb[((size_t)b * 36 + 4 * a + d) * HWPX + pos];
      float diff = pred - tg[d];
      float ad = fabsf(diff);
      sm += (ad < (1.0f / 9.0f)) ? 4.5f * diff * diff : ad - (0.5f / 9.0f);
    }
    atomicAdd(&acc[1 + b], sm);
  }
}

// ---------------- kernel 11: finalize losses ----------------

__global__ void finalize(const float* __restrict__ acc,
                         const int* __restrict__ cnt,
                         float* __restrict__ out) {
  if (blockIdx.x != 0 || threadIdx.x != 0) return;
  int tot = 0;
  for (int b = 0; b < BN; ++b) tot += cnt[b];
  int dt = tot > 1 ? tot : 1;
  out[BN * POST_NMS_ * 5 + 0] = acc[0] / (float)dt;
  float lb = 0.0f;
  for (int b = 0; b < BN; ++b) {
    int cb = cnt[b] > 1 ? cnt[b] : 1;
    lb += acc[1 + b] / (float)cb;
  }
  out[BN * POST_NMS_ * 5 + 1] = lb / (float)BN;
}

// ---------------- host launcher ----------------

extern "C" void kernel_launch(void* const* d_in, const int* in_sizes, int n_in,
                              void* d_out, int out_size, void* d_ws, size_t ws_size,
                              hipStream_t stream) {
  const float* base_feat = (const float*)d_in[0];
  const float* im_info   = (const float*)d_in[1];
  const float* gt_boxes  = (const float*)d_in[2];
  const float* w_pre     = (const float*)d_in[3];
  const float* b_pre     = (const float*)d_in[4];
  const float* w_cls     = (const float*)d_in[5];
  const float* b_cls     = (const float*)d_in[6];
  const float* w_box     = (const float*)d_in[7];
  const float* b_box     = (const float*)d_in[8];
  float* out = (float*)d_out;

  char* base = (char*)d_ws;
  size_t off = 0;
  auto carve = [&](size_t bytes) -> char* {
    char* p = base + off;
    off = (off + bytes + 255) & ~(size_t)255;
    return p;
  };
  float* feat      = (float*)carve((size_t)BN * OUTC_ * HWPX * 4);
  float* scoresb   = (float*)carve((size_t)BN * 18 * HWPX * 4);
  float* deltasb   = (float*)carve((size_t)BN * 36 * HWPX * 4);
  float* pbox      = (float*)carve((size_t)BN * NANCH * 4 * 4);
  float* pscore    = (float*)carve((size_t)BN * NANCH * 4);
  unsigned long long* keys = (unsigned long long*)carve((size_t)BN * NPAD * 8);
  float* tbox      = (float*)carve((size_t)BN * PRE_NMS_ * 4 * 4);
  float* tsc       = (float*)carve((size_t)BN * PRE_NMS_ * 4);
  float* maxov     = (float*)carve((size_t)BN * NANCH * 4);
  int*   argmx     = (int*)carve((size_t)BN * NANCH * 4);
  int*   labels    = (int*)carve((size_t)BN * NANCH * 4);
  unsigned* gtmax  = (unsigned*)carve((size_t)BN * 20 * 4);
  float* acc       = (float*)carve((size_t)(1 + BN) * 4);
  int*   cnt       = (int*)carve((size_t)BN * 4);

  // 1. 3x3 conv via WMMA: 4 batches * 16 mtiles * 192 ntiles / 8 waves = 1536 blocks
  conv3x3_wmma<<<1536, 256, 0, stream>>>(base_feat, w_pre, b_pre, feat);
  // 2. heads
  {
    int tot = BN * 54 * HWPX;
    conv1x1_heads<<<(tot + 255) / 256, 256, 0, stream>>>(
        feat, w_cls, b_cls, w_box, b_box, scoresb, deltasb);
  }
  // 3. proposals + keys
  {
    int tot = BN * NPAD;
    proposals<<<(tot + 255) / 256, 256, 0, stream>>>(
        scoresb, deltasb, im_info, pbox, pscore, keys);
  }
  // 4. per-image sort
  bitonic_sort<<<BN, 1024, 0, stream>>>(keys);
  // 5. NMS + rois
  nms_rois<<<BN, 256, 0, stream>>>(keys, pbox, pscore, tbox, tsc, out);
  // 6. init accumulators
  init_acc<<<1, 128, 0, stream>>>(gtmax, acc, cnt);
  // 7. IoU stats
  {
    int tot = BN * NANCH;
    iou_stats<<<(tot + 255) / 256, 256, 0, stream>>>(
        gt_boxes, im_info, maxov, argmx, gtmax);
    // 8. labels
    assign_labels<<<(tot + 255) / 256, 256, 0, stream>>>(
        gt_boxes, im_info, maxov, gtmax, labels);
  }
  // 9. subsample
  subsample<<<BN, 1024, 0, stream>>>(labels);
  // 10. losses
  {
    int tot = BN * NANCH;
    loss_accum<<<(tot + 255) / 256, 256, 0, stream>>>(
        scoresb, deltasb, gt_boxes, labels, argmx, acc, cnt);
  }
  // 11. finalize
  finalize<<<1, 1, 0, stream>>>(acc, cnt, out);
  (void)in_sizes; (void)n_in; (void)out_size; (void)ws_size;
}